// KVCache_77429670412928
// MI455X (gfx1250) — compile-verified
//
#include <hip/hip_runtime.h>

// ---------------------------------------------------------------------------
// KV-cache update for MI455X (gfx1250). Pure data movement: ~515 MB traffic,
// 0 FLOPs -> HBM-bandwidth bound (~22us floor @ 23.3 TB/s). Strategy:
//   * 128-bit transfers (each access = fully coalesced 512B per wave32)
//   * 4 transfers in flight per wave before one s_wait_asynccnt (MLP)
//   * k-vs-v selection via blockIdx.y (uniform SGPR select, no lane cndmasks)
//   * 32-bit offsets -> saddr+voffset addressing, minimal VALU
//   * gfx1250 async Global->LDS->Global engine (ASYNCcnt) where exposed
// ---------------------------------------------------------------------------

typedef float f4 __attribute__((ext_vector_type(4)));
typedef int   i4 __attribute__((ext_vector_type(4)));
// Exact element type the async-LDS builtins expect: int __vector_size__(16)
typedef int   avi4 __attribute__((vector_size(16)));

// Problem dims (match reference)
constexpr int B_ = 2, H_ = 32, L_ = 4096, D_ = 128, T_ = 2048, G_ = 4;
constexpr int D4_ = D_ / 4;                                       // 32 float4 / row
constexpr long long CACHE_ELEMS = (long long)B_ * H_ * L_ * D_;   // 33,554,432 floats
constexpr unsigned PRE4  = (unsigned)B_ * H_ * T_ * D4_;          // 4,194,304 float4 / cache
constexpr unsigned TAIL4 = (unsigned)B_ * H_ * (L_ - T_) * D4_;   // 4,194,304 float4 / cache
constexpr unsigned POS4  = (unsigned)B_ * H_ * L_ / 4;            // 65,536 int4

constexpr int PT = 4;   // float4 elements copied per thread (MLP per wave)

#if defined(__gfx1250__) &&                                          \
    __has_builtin(__builtin_amdgcn_global_load_async_to_lds_b128) && \
    __has_builtin(__builtin_amdgcn_global_store_async_from_lds_b128) && \
    __has_builtin(__builtin_amdgcn_s_wait_asynccnt)
#define USE_ASYNC_LDS 1
typedef __attribute__((address_space(1))) avi4 g_avi4;
typedef __attribute__((address_space(3))) avi4 l_avi4;
#endif

// ---- Kernel 1: prefill copy  k_val/v_val -> cache rows [0, T) -------------
// Grid: (PRE4/(256*PT), 2); blockIdx.y selects K (0) or V (1) cache.
__global__ __launch_bounds__(256) void prefill_copy(
    const f4* __restrict__ k_val, const f4* __restrict__ v_val,
    f4* __restrict__ out_k, f4* __restrict__ out_v) {
  const f4* __restrict__ src = blockIdx.y ? v_val : k_val;   // uniform (SGPR select)
  f4* __restrict__ dst       = blockIdx.y ? out_v : out_k;
  unsigned base = blockIdx.x * (256u * PT) + threadIdx.x;

#if defined(USE_ASYNC_LDS)
  __shared__ f4 buf[256 * PT];
#pragma unroll
  for (int k = 0; k < PT; ++k) {               // ASYNCcnt builds to PT
    unsigned i = base + k * 256u;              // flat src float4 index
    __builtin_amdgcn_global_load_async_to_lds_b128(
        (g_avi4*)(src + i), (l_avi4*)&buf[threadIdx.x + k * 256], 0, 0);
  }
  __builtin_amdgcn_s_wait_asynccnt(0);         // lane-private LDS slots
#pragma unroll
  for (int k = 0; k < PT; ++k) {
    unsigned i = base + k * 256u;
    unsigned srow = i >> 5, d4 = i & 31u;      // srow = bh*T + l
    unsigned bh = srow >> 11, l = srow & (T_ - 1u);
    unsigned di = (((bh << 12) | l) << 5) + d4;  // (bh*L + l)*32 + d4
    __builtin_amdgcn_global_store_async_from_lds_b128(
        (g_avi4*)(dst + di), (l_avi4*)&buf[threadIdx.x + k * 256], 0, 0);
  }
  // S_ENDPGM performs an implicit wait-idle, draining ASYNCcnt.
#else
  f4 v[PT];
#pragma unroll
  for (int k = 0; k < PT; ++k)
    v[k] = __builtin_nontemporal_load(src + (base + k * 256u));
#pragma unroll
  for (int k = 0; k < PT; ++k) {
    unsigned i = base + k * 256u;
    unsigned srow = i >> 5, d4 = i & 31u;
    unsigned bh = srow >> 11, l = srow & (T_ - 1u);
    unsigned di = (((bh << 12) | l) << 5) + d4;
    __builtin_nontemporal_store(v[k], dst + di);
  }
#endif
}

// ---- Kernel 2: tail passthrough  cache rows [T, L) ------------------------
// Grid: (TAIL4/(256*PT), 2); blockIdx.y selects K (0) or V (1) cache.
__global__ __launch_bounds__(256) void tail_copy(
    const f4* __restrict__ kc_in, const f4* __restrict__ vc_in,
    f4* __restrict__ out_k, f4* __restrict__ out_v) {
  const f4* __restrict__ src = blockIdx.y ? vc_in : kc_in;   // uniform (SGPR select)
  f4* __restrict__ dst       = blockIdx.y ? out_v : out_k;
  unsigned base = blockIdx.x * (256u * PT) + threadIdx.x;

#if defined(USE_ASYNC_LDS)
  __shared__ f4 buf[256 * PT];
#pragma unroll
  for (int k = 0; k < PT; ++k) {
    unsigned j = base + k * 256u;
    unsigned trow = j >> 5, d4 = j & 31u;      // trow = bh*(L-T) + lt
    unsigned bh = trow >> 11, lt = trow & (L_ - T_ - 1u);
    unsigned x = (((bh << 12) | (T_ + lt)) << 5) + d4;   // same layout in & out
    __builtin_amdgcn_global_load_async_to_lds_b128(
        (g_avi4*)(src + x), (l_avi4*)&buf[threadIdx.x + k * 256], 0, 0);
  }
  __builtin_amdgcn_s_wait_asynccnt(0);
#pragma unroll
  for (int k = 0; k < PT; ++k) {
    unsigned j = base + k * 256u;
    unsigned trow = j >> 5, d4 = j & 31u;
    unsigned bh = trow >> 11, lt = trow & (L_ - T_ - 1u);
    unsigned x = (((bh << 12) | (T_ + lt)) << 5) + d4;
    __builtin_amdgcn_global_store_async_from_lds_b128(
        (g_avi4*)(dst + x), (l_avi4*)&buf[threadIdx.x + k * 256], 0, 0);
  }
#else
  f4 v[PT];
  unsigned x[PT];
#pragma unroll
  for (int k = 0; k < PT; ++k) {
    unsigned j = base + k * 256u;
    unsigned trow = j >> 5, d4 = j & 31u;
    unsigned bh = trow >> 11, lt = trow & (L_ - T_ - 1u);
    x[k] = (((bh << 12) | (T_ + lt)) << 5) + d4;
    v[k] = __builtin_nontemporal_load(src + x[k]);
  }
#pragma unroll
  for (int k = 0; k < PT; ++k)
    __builtin_nontemporal_store(v[k], dst + x[k]);
#endif
}

// ---- Kernel 3: pos tensor -------------------------------------------------
// pos[b,h,l] = L if l<G; input_pos[l] if l<T; else pos_in[b,h,l]
__global__ __launch_bounds__(256) void pos_fill(
    const i4* __restrict__ pos_in, const int* __restrict__ input_pos,
    i4* __restrict__ out_pos) {
  unsigned idx = blockIdx.x * 256u + threadIdx.x;            // int4 index, [0, POS4)
  unsigned l0 = (idx & (L_ / 4u - 1u)) << 2;                 // first l of this int4
  i4 r;
  if (l0 >= (unsigned)T_) {                                  // tail: passthrough (-1s)
    r = __builtin_nontemporal_load(pos_in + idx);
  } else {                                                   // prefill region
    r = ((const i4*)input_pos)[l0 >> 2];
    if (l0 < (unsigned)G_) {
#pragma unroll
      for (int j = 0; j < 4; ++j)
        if (l0 + j < (unsigned)G_) r[j] = L_;                // global tokens -> priority L
    }
  }
  __builtin_nontemporal_store(r, out_pos + idx);
}

// ---- Kernel 4: decode-step per-head scatter (runs AFTER fills) ------------
__global__ __launch_bounds__(256) void scatter_step(
    const f4* __restrict__ k_step, const f4* __restrict__ v_step,
    const int* __restrict__ fill_indices, const int* __restrict__ input_pos_step,
    f4* __restrict__ out_k, f4* __restrict__ out_v, int* __restrict__ out_pos) {
  unsigned tid = blockIdx.x * 256u + threadIdx.x;   // [0, B*H*D4) = 2048
  if (tid >= (unsigned)(B_ * H_ * D4_)) return;
  unsigned d4 = tid & 31u;
  unsigned bh = tid >> 5;                           // b*H + h
  unsigned h = bh & (H_ - 1u);
  unsigned slot = (unsigned)fill_indices[h];
  unsigned dst = (bh * L_ + slot) * D4_ + d4;       // < 8.4M, fits u32
  unsigned src = bh * D4_ + d4;
  out_k[dst] = k_step[src];
  out_v[dst] = v_step[src];
  if (d4 == 0) out_pos[bh * L_ + slot] = input_pos_step[0];
}

// ---------------------------------------------------------------------------
extern "C" void kernel_launch(void* const* d_in, const int* in_sizes, int n_in,
                              void* d_out, int out_size, void* d_ws, size_t ws_size,
                              hipStream_t stream) {
  (void)in_sizes; (void)n_in; (void)out_size; (void)d_ws; (void)ws_size;

  const f4*  kc_in        = (const f4*)d_in[0];   // k_cache  (B,H,L,D) f32
  const f4*  vc_in        = (const f4*)d_in[1];   // v_cache
  const i4*  pos_in       = (const i4*)d_in[2];   // pos      (B,H,L) i32
  const int* input_pos    = (const int*)d_in[3];  // (T,) i32
  const f4*  k_val        = (const f4*)d_in[4];   // (B,H,T,D) f32
  const f4*  v_val        = (const f4*)d_in[5];
  const int* fill_indices = (const int*)d_in[6];  // (H,) i32
  const f4*  k_step       = (const f4*)d_in[7];   // (B,H,1,D) f32
  const f4*  v_step       = (const f4*)d_in[8];
  const int* ip_step      = (const int*)d_in[9];  // scalar i32

  float* out   = (float*)d_out;
  f4*  out_k   = (f4*)out;
  f4*  out_v   = (f4*)(out + CACHE_ELEMS);
  int* out_pos = (int*)(out + 2 * CACHE_ELEMS);

  // Streaming fills (disjoint output regions; stream order only matters for scatter)
  dim3 gpre(PRE4 / (256 * PT), 2);    // (4096, 2): y selects K/V
  dim3 gtail(TAIL4 / (256 * PT), 2);  // (4096, 2)
  prefill_copy<<<gpre, 256, 0, stream>>>(k_val, v_val, out_k, out_v);
  tail_copy  <<<gtail, 256, 0, stream>>>(kc_in, vc_in, out_k, out_v);
  pos_fill   <<<POS4 / 256, 256, 0, stream>>>(pos_in, input_pos, (i4*)out_pos);
  // Decode-step scatter must override the fills: same stream => ordered after.
  scatter_step<<<(B_ * H_ * D4_ + 255) / 256, 256, 0, stream>>>(
      k_step, v_step, fill_indices, ip_step, out_k, out_v, out_pos);
}